// GNN_24807731102420
// MI455X (gfx1250) — compile-verified
//
#include <hip/hip_runtime.h>
#include <math.h>

// ---------------- problem constants (match reference) ----------------
#define NNODES 50000
#define NEDGES 800000
#define ETOT   (NEDGES + NNODES)   // edges + self loops = 850000
#define F_IN   128
#define HID    64
#define CLS    40
#define NEG_SLOPE 0.2f
#define EPSV 1e-16f

typedef float v2f __attribute__((ext_vector_type(2)));
typedef float v8f __attribute__((ext_vector_type(8)));

// ---------------------------------------------------------------------
// WMMA fp32 GEMM: H[N x Fout] = X[N x Fin] * W[Fin x Fout]
// One wave computes one 16x16 output tile via V_WMMA_F32_16X16X4_F32.
// blockDim.x = 32 * ctiles (ctiles 16-col slabs), gridDim.x = N/16.
// A layout (32-bit 16x4): lanes 0-15 -> M=lane, K={k,k+1}; lanes 16-31 -> K={k+2,k+3}
// B layout (32-bit 4x16): lanes 0-15 -> N=lane, K={k,k+1}; lanes 16-31 -> K={k+2,k+3}
// C/D layout: VGPR i: lanes 0-15 -> M=i, lanes 16-31 -> M=i+8, N=lane&15
// Out-of-range columns (layer 3: Fout=40 padded to 48) are handled branch-free
// by zero-masking B so EXEC stays all-1s through every WMMA (ISA §7.12 req).
// ---------------------------------------------------------------------
__global__ void gat_gemm_wmma(const float* __restrict__ X, const float* __restrict__ W,
                              float* __restrict__ H, int Fin, int Fout) {
    const int lane = threadIdx.x & 31;
    const int wave = threadIdx.x >> 5;
    const int l    = lane & 15;
    const int hh   = lane >> 4;           // half-wave select
    const int rowBase = blockIdx.x << 4;  // N is an exact multiple of 16
    const int col  = (wave << 4) + l;
    const int colL = (col < Fout) ? col : (Fout - 1);
    const float bmask = (col < Fout) ? 1.0f : 0.0f;

    const float* xrow = X + (size_t)(rowBase + l) * Fin;

    v8f c = {0.f, 0.f, 0.f, 0.f, 0.f, 0.f, 0.f, 0.f};
    for (int k = 0; k < Fin; k += 4) {
        const int ka = k + (hh << 1);
        v2f a, b;
        a.x = xrow[ka];
        a.y = xrow[ka + 1];
        b.x = W[(size_t)ka * Fout + colL] * bmask;
        b.y = W[(size_t)(ka + 1) * Fout + colL] * bmask;
        // (neg_a, A, neg_b, B, c_mod, C, reuse_a, reuse_b)
        c = __builtin_amdgcn_wmma_f32_16x16x4_f32(false, a, false, b, (short)0, c,
                                                  false, false);
    }

    if (col < Fout) {
        const int orow = rowBase + (hh << 3);
        for (int i = 0; i < 8; ++i)
            H[(size_t)(orow + i) * Fout + col] = c[i];
    }
}

// ---------------- edge prep: int32 src/dst with self loops -----------
__global__ void gat_prep_edges(const int* __restrict__ edge_index,
                               int* __restrict__ srcA, int* __restrict__ dstA) {
    int i = blockIdx.x * blockDim.x + threadIdx.x;
    if (i >= ETOT) return;
    if (i < NEDGES) {
        srcA[i] = edge_index[i];
        dstA[i] = edge_index[NEDGES + i];
    } else {
        srcA[i] = i - NEDGES;   // self loop
        dstA[i] = i - NEDGES;
    }
}

// ---------------- per-node attention dots ----------------------------
__global__ void gat_dots(const float* __restrict__ h,
                         const float* __restrict__ a_src, const float* __restrict__ a_dst,
                         float* __restrict__ ssrc, float* __restrict__ sdst, int F) {
    int i = blockIdx.x * blockDim.x + threadIdx.x;
    if (i >= NNODES) return;
    const float* row = h + (size_t)i * F;
    float as = 0.f, ad = 0.f;
    for (int f = 0; f < F; ++f) {
        float v = row[f];
        as += v * a_src[f];
        ad += v * a_dst[f];
    }
    ssrc[i] = as;
    sdst[i] = ad;
}

// ---------------- init: m=-inf, denom=0, agg=bias --------------------
__global__ void gat_init(float* __restrict__ m, float* __restrict__ denom,
                         float* __restrict__ agg, const float* __restrict__ bias, int F) {
    int i = blockIdx.x * blockDim.x + threadIdx.x;
    int total = NNODES * F;
    if (i < total) agg[i] = bias[i % F];
    if (i < NNODES) {
        m[i] = -INFINITY;
        denom[i] = 0.f;
    }
}

// ---------------- float atomic max (sign-split int trick) ------------
__device__ __forceinline__ void atomicMaxF(float* addr, float val) {
    if (val >= 0.f) atomicMax((int*)addr, __float_as_int(val));
    else            atomicMin((unsigned int*)addr, __float_as_uint(val));
}

// ---------------- edge score + segment max ---------------------------
__global__ void gat_edge_max(const int* __restrict__ srcA, const int* __restrict__ dstA,
                             const float* __restrict__ ssrc, const float* __restrict__ sdst,
                             float* __restrict__ escore, float* __restrict__ m) {
    int i = blockIdx.x * blockDim.x + threadIdx.x;
    if (i >= ETOT) return;
    int s = srcA[i], d = dstA[i];
    float v = ssrc[s] + sdst[d];
    float lr = (v > 0.f) ? v : NEG_SLOPE * v;   // leaky_relu
    escore[i] = lr;
    atomicMaxF(&m[d], lr);
}

// ---------------- exp(e - m[dst]) + segment sum ----------------------
__global__ void gat_edge_exp(const int* __restrict__ dstA,
                             float* __restrict__ escore,
                             const float* __restrict__ m, float* __restrict__ denom) {
    int i = blockIdx.x * blockDim.x + threadIdx.x;
    if (i >= ETOT) return;
    int d = dstA[i];
    float ex = expf(escore[i] - m[d]);
    escore[i] = ex;
    atomicAdd(&denom[d], ex);
}

// ---------------- weighted scatter: agg[dst] += h[src] * alpha -------
// 64 lanes per edge (f = tid & 63), guarded for F < 64.
__global__ void gat_aggregate(const int* __restrict__ srcA, const int* __restrict__ dstA,
                              const float* __restrict__ escore, const float* __restrict__ denom,
                              const float* __restrict__ h, float* __restrict__ agg, int F) {
    long long i = (long long)blockIdx.x * blockDim.x + threadIdx.x;
    long long e = i >> 6;
    int f = (int)(i & 63);
    if (e >= ETOT || f >= F) return;
    int s = srcA[e], d = dstA[e];
    float alpha = escore[e] / (denom[d] + EPSV);
    atomicAdd(&agg[(size_t)d * F + f], h[(size_t)s * F + f] * alpha);
}

// ---------------- exact-erf GELU (in place) --------------------------
__global__ void gat_gelu(float* __restrict__ x, int total) {
    int i = blockIdx.x * blockDim.x + threadIdx.x;
    if (i >= total) return;
    float v = x[i];
    x[i] = 0.5f * v * (1.0f + erff(v * 0.70710678118654752f));
}

// ---------------------------------------------------------------------
// host-side layer driver (all launches on `stream`; stream order makes
// in-place buffer reuse across layers safe)
// ---------------------------------------------------------------------
static void run_gat_layer(const float* x, int Fin, int Fout, int ctiles,
                          const float* W, const float* asrc, const float* adst,
                          const float* bias,
                          float* h, float* agg, bool do_gelu,
                          float* ssrc, float* sdst, float* m, float* denom,
                          float* escore, const int* srcA, const int* dstA,
                          hipStream_t stream) {
    const int B = 256;
    gat_gemm_wmma<<<NNODES / 16, 32 * ctiles, 0, stream>>>(x, W, h, Fin, Fout);
    gat_dots<<<(NNODES + B - 1) / B, B, 0, stream>>>(h, asrc, adst, ssrc, sdst, Fout);
    gat_init<<<((size_t)NNODES * Fout + B - 1) / B, B, 0, stream>>>(m, denom, agg, bias, Fout);
    gat_edge_max<<<(ETOT + B - 1) / B, B, 0, stream>>>(srcA, dstA, ssrc, sdst, escore, m);
    gat_edge_exp<<<(ETOT + B - 1) / B, B, 0, stream>>>(dstA, escore, m, denom);
    long long aggThreads = (long long)ETOT * 64;
    gat_aggregate<<<(unsigned)((aggThreads + B - 1) / B), B, 0, stream>>>(
        srcA, dstA, escore, denom, h, agg, Fout);
    if (do_gelu)
        gat_gelu<<<((size_t)NNODES * Fout + B - 1) / B, B, 0, stream>>>(agg, NNODES * Fout);
}

extern "C" void kernel_launch(void* const* d_in, const int* in_sizes, int n_in,
                              void* d_out, int out_size, void* d_ws, size_t ws_size,
                              hipStream_t stream) {
    // inputs in setup_inputs() order
    const float* x      = (const float*)d_in[0];
    const int*   eidx   = (const int*)d_in[1];
    const float* W1     = (const float*)d_in[2];
    const float* as1    = (const float*)d_in[3];
    const float* ad1    = (const float*)d_in[4];
    const float* b1     = (const float*)d_in[5];
    const float* W2     = (const float*)d_in[6];
    const float* as2    = (const float*)d_in[7];
    const float* ad2    = (const float*)d_in[8];
    const float* b2     = (const float*)d_in[9];
    const float* W3     = (const float*)d_in[10];
    const float* as3    = (const float*)d_in[11];
    const float* ad3    = (const float*)d_in[12];
    const float* b3     = (const float*)d_in[13];
    float* out = (float*)d_out;

    // carve workspace (~37 MB)
    char* p = (char*)d_ws;
    auto carve = [&](size_t bytes) -> void* {
        void* r = (void*)p;
        p += (bytes + 255) & ~(size_t)255;
        return r;
    };
    float* hbuf   = (float*)carve((size_t)NNODES * HID * 4);
    float* agg    = (float*)carve((size_t)NNODES * HID * 4);
    float* ssrc   = (float*)carve((size_t)NNODES * 4);
    float* sdst   = (float*)carve((size_t)NNODES * 4);
    float* m      = (float*)carve((size_t)NNODES * 4);
    float* denom  = (float*)carve((size_t)NNODES * 4);
    float* escore = (float*)carve((size_t)ETOT * 4);
    int*   srcA   = (int*)carve((size_t)ETOT * 4);
    int*   dstA   = (int*)carve((size_t)ETOT * 4);
    (void)ws_size; (void)in_sizes; (void)n_in; (void)out_size;

    gat_prep_edges<<<(ETOT + 255) / 256, 256, 0, stream>>>(eidx, srcA, dstA);

    // layer 1: 128 -> 64, GELU        (x -> hbuf -> agg)
    run_gat_layer(x,   F_IN, HID, 4, W1, as1, ad1, b1, hbuf, agg, true,
                  ssrc, sdst, m, denom, escore, srcA, dstA, stream);
    // layer 2: 64 -> 64, GELU         (agg -> hbuf -> agg; stream order makes reuse safe)
    run_gat_layer(agg, HID,  HID, 4, W2, as2, ad2, b2, hbuf, agg, true,
                  ssrc, sdst, m, denom, escore, srcA, dstA, stream);
    // layer 3: 64 -> 40, no GELU      (agg -> hbuf -> d_out), 40 cols padded to 3 tiles
    run_gat_layer(agg, HID,  CLS, 3, W3, as3, ad3, b3, hbuf, out, false,
                  ssrc, sdst, m, denom, escore, srcA, dstA, stream);
}